// GraphConv_27255862461193
// MI455X (gfx1250) — compile-verified
//
#include <hip/hip_runtime.h>
#include <hip/hip_bf16.h>

typedef __attribute__((ext_vector_type(2))) float v2f;
typedef __attribute__((ext_vector_type(8))) float v8f;
typedef __attribute__((ext_vector_type(4))) unsigned int u32x4;
typedef __attribute__((ext_vector_type(8))) int i32x8;
typedef __attribute__((ext_vector_type(4))) int i32x4;

#define WMMA_F32(a, b, c) \
    __builtin_amdgcn_wmma_f32_16x16x4_f32(false, (a), false, (b), (short)0, (c), false, false)

#if defined(__has_builtin)
#if __has_builtin(__builtin_amdgcn_tensor_load_to_lds) && __has_builtin(__builtin_amdgcn_s_wait_tensorcnt)
#define HAVE_TDM 1
#endif
#endif
#ifndef HAVE_TDM
#define HAVE_TDM 0
#endif

// Problem constants (from reference)
constexpr int Bsz = 32, Cc = 64, Tt = 12, Nn = 1024, KSu = 3, OUTc = 64;
constexpr int TN = 128;  // n-columns per workgroup (8 col-tiles of 16)
constexpr int S  = 66;   // LDS row stride for 64-float rows (= TDM pad: 64 DW data + 2 DW pad)
constexpr int SO = 130;  // LDS row stride for 128-wide rows (Ys / x2 tile)

// LDS layout (floats):
//   Th  [3][64][S]   : theta transposed ThT[k][o][i]   (dead after GEMM2; x2 tile aliases)
//   Fw  [64][S]      : fc_w[oo][c]                     (FUSE_FC only)
//   Xs0/Bs0, Xs1/Bs1 : double-buffered X chunk [64][S] and Lk chunk [3][128][S]
//   Ys  [3][64][SO]  : aliases Bs0 (24960 <= 25344)
//   Outs[64][SO]     : aliases Th  (8320  <= 12672)
constexpr int TH_OFF  = 0;                         // 12672
constexpr int FW_OFF  = TH_OFF + KSu * 64 * S;     // + 4224
constexpr int XS0_OFF = FW_OFF + 64 * S;           // + 4224
constexpr int BS0_OFF = XS0_OFF + 64 * S;          // + 25344
constexpr int XS1_OFF = BS0_OFF + KSu * 128 * S;   // + 4224
constexpr int BS1_OFF = XS1_OFF + 64 * S;          // + 25344
constexpr int LDS_FLOATS = BS1_OFF + KSu * 128 * S; // 76032 floats = 297 KB

#if HAVE_TDM
// ---- Tensor Data Mover: 2D tile load, element size 4B, LDS pad 2 DW per 64 DW (stride 66)
__device__ __forceinline__ void tdm_load_2d(unsigned lds_byte_addr, const void* gptr,
                                            unsigned tile_d0, unsigned tile_d1,
                                            unsigned long long stride0_elts) {
    unsigned long long ga = (unsigned long long)gptr;
    u32x4 g0;
    g0[0] = 1u;                                              // count=1 (valid user D#)
    g0[1] = lds_byte_addr;                                   // lds_addr
    g0[2] = (unsigned)(ga & 0xFFFFFFFFu);                    // global_addr[31:0]
    g0[3] = (unsigned)((ga >> 32) & 0x1FFFFFFu) | (2u << 30); // global_addr[56:32] | type=2
    i32x8 g1;
    // workgroup_mask=0 | data_size=2(4B) | pad_enable | pad_interval=5(64DW) | pad_amount=1(2DW)
    g1[0] = (int)((2u << 16) | (1u << 20) | (5u << 22) | (1u << 25));
    g1[1] = (int)((tile_d0 & 0xFFFFu) << 16);                // abar_addr=0 | tensor_dim0[15:0]=tile_d0
    g1[2] = (int)(((tile_d0 >> 16) & 0xFFFFu) | ((tile_d1 & 0xFFFFu) << 16)); // dim0 hi | tensor_dim1 lo
    g1[3] = (int)(((tile_d1 >> 16) & 0xFFFFu) | ((tile_d0 & 0xFFFFu) << 16)); // dim1 hi | tile_dim0
    g1[4] = (int)(tile_d1 & 0xFFFFu);                        // tile_dim1 | tile_dim2=0 (2D)
    g1[5] = (int)(stride0_elts & 0xFFFFFFFFu);               // tensor_dim0_stride[31:0]
    g1[6] = (int)((stride0_elts >> 32) & 0xFFFFu);           // stride hi | dim1_stride lo = 0
    g1[7] = 0;
    i32x4 g2 = {0, 0, 0, 0};
    i32x4 g3 = {0, 0, 0, 0};
#if __clang_major__ >= 23
    i32x8 g4 = {0, 0, 0, 0, 0, 0, 0, 0};
    __builtin_amdgcn_tensor_load_to_lds(g0, g1, g2, g3, g4, 0);
#else
    __builtin_amdgcn_tensor_load_to_lds(g0, g1, g2, g3, 0);
#endif
}

// issue one m-chunk: X tile [64 x 64] + 3 Lk tiles [128 x 64]
__device__ __forceinline__ void issue_chunk(const float* xb, const float* Lk,
                                            int n0, int m0,
                                            unsigned lds_base, int xs_off, int bs_off) {
    tdm_load_2d(lds_base + (unsigned)xs_off * 4u, xb + m0, 64u, 64u,
                (unsigned long long)(Tt * Nn));
#pragma unroll
    for (int k = 0; k < KSu; ++k) {
        tdm_load_2d(lds_base + (unsigned)(bs_off + k * 128 * S) * 4u,
                    Lk + ((size_t)k * Nn + n0) * Nn + m0, 64u, 128u,
                    (unsigned long long)Nn);
    }
}
#endif // HAVE_TDM

// one 64-deep K-chunk of GEMM1 from LDS buffers
__device__ __forceinline__ void compute_chunk(const float* __restrict__ smem,
                                              int xs_off, int bs_off,
                                              int tr2, int tc2, int l15, int half,
                                              v8f (&acc)[KSu][2][2]) {
    const float* Ar0 = smem + xs_off + ((tr2 + 0) * 16 + l15) * S;
    const float* Ar1 = smem + xs_off + ((tr2 + 1) * 16 + l15) * S;
    for (int kb = 0; kb < 64; kb += 4) {
        v2f a0 = *(const v2f*)(Ar0 + kb + 2 * half);
        v2f a1 = *(const v2f*)(Ar1 + kb + 2 * half);
#pragma unroll
        for (int k = 0; k < KSu; ++k) {
            const float* Bk = smem + bs_off + k * 128 * S;
            v2f b0 = *(const v2f*)(Bk + ((tc2 + 0) * 16 + l15) * S + kb + 2 * half);
            v2f b1 = *(const v2f*)(Bk + ((tc2 + 1) * 16 + l15) * S + kb + 2 * half);
            acc[k][0][0] = WMMA_F32(a0, b0, acc[k][0][0]);
            acc[k][0][1] = WMMA_F32(a0, b1, acc[k][0][1]);
            acc[k][1][0] = WMMA_F32(a1, b0, acc[k][1][0]);
            acc[k][1][1] = WMMA_F32(a1, b1, acc[k][1][1]);
        }
    }
}

template <bool FUSE_FC>
__global__ __launch_bounds__(256)
void gconv_layer_kernel(const float* __restrict__ x_in,
                        const float* __restrict__ Lk,
                        const float* __restrict__ theta,
                        const float* __restrict__ bias,
                        const float* __restrict__ fc_w,
                        const float* __restrict__ fc_b,
                        float* __restrict__ out) {
    extern __shared__ float smem[];

    const int tid  = threadIdx.x;
    const int wave = tid >> 5;
    const int lane = tid & 31;
    const int l15  = lane & 15;
    const int half = lane >> 4;

    // wave -> 2x2 block of 16x16 tiles inside the 4(row) x 8(col) tile grid
    const int tr2 = (wave & 1) * 2;   // row tiles (i / o dim, 64)
    const int tc2 = (wave >> 1) * 2;  // col tiles (nn dim, 128)

    const int n0 = blockIdx.x * TN;
    const int bt = blockIdx.y;
    const int b  = bt / Tt;
    const int t  = bt - b * Tt;

    const float* xb = x_in + (size_t)b * Cc * Tt * Nn + (size_t)t * Nn; // xb[i*T*N + m]

    // ---- stage theta (transposed): ThT[k][o][i] = theta[(i*C+o)*KS+k]
    {
        float* Th = smem + TH_OFF;
        for (int e = tid; e < Cc * Cc * KSu; e += 256) {
            int i   = e / (Cc * KSu);
            int rem = e - i * (Cc * KSu);
            int o   = rem / KSu;
            int k   = rem - o * KSu;
            Th[(k * 64 + o) * S + i] = theta[e];
        }
    }
    // ---- stage fc_w: Fw[oo][c]
    if (FUSE_FC) {
        float* Fw = smem + FW_OFF;
        for (int e = tid; e < OUTc * Cc; e += 256) {
            Fw[(e >> 6) * S + (e & 63)] = fc_w[e];
        }
    }

    // ---- GEMM1: Y_k[i, nn] = sum_m X[b,i,t,m] * Lk[k, n0+nn, m]
    v8f acc[KSu][2][2];
    {
        v8f z = {};
#pragma unroll
        for (int k = 0; k < KSu; ++k)
#pragma unroll
            for (int ri = 0; ri < 2; ++ri) {
                acc[k][ri][0] = z;
                acc[k][ri][1] = z;
            }
    }

#if HAVE_TDM
    // ---- double-buffered TDM pipeline: wave 0 drives the Tensor Data Mover
    const unsigned lds_base = (unsigned)(unsigned long long)(void*)smem;
    const bool issuer = (wave == 0);
    if (issuer) issue_chunk(xb, Lk, n0, 0, lds_base, XS0_OFF, BS0_OFF);
    for (int mc = 0; mc < Nn / 64; ++mc) {
        const int cur = mc & 1;
        if (issuer) __builtin_amdgcn_s_wait_tensorcnt(0);   // chunk mc landed in LDS
        __syncthreads();                                    // data visible; prev buffer free
        if (issuer && mc + 1 < Nn / 64) {
            issue_chunk(xb, Lk, n0, (mc + 1) * 64, lds_base,
                        cur ? XS0_OFF : XS1_OFF, cur ? BS0_OFF : BS1_OFF);
        }
        compute_chunk(smem, cur ? XS1_OFF : XS0_OFF, cur ? BS1_OFF : BS0_OFF,
                      tr2, tc2, l15, half, acc);
    }
    __syncthreads();
#else
    // ---- fallback: cooperative copy, single buffer
    for (int mc = 0; mc < Nn / 64; ++mc) {
        const int m0  = mc * 64;
        const int row = tid >> 4;
        const int c4  = (tid & 15) * 4;
#pragma unroll
        for (int rr = 0; rr < 4; ++rr) {
            int i = row + rr * 16;
            float4 v = *(const float4*)(xb + (size_t)i * Tt * Nn + m0 + c4);
            float* dst = smem + XS0_OFF + i * S + c4;
            dst[0] = v.x; dst[1] = v.y; dst[2] = v.z; dst[3] = v.w;
        }
#pragma unroll
        for (int rr = 0; rr < 24; ++rr) {
            int r  = row + rr * 16;
            int k  = r >> 7;
            int nn = r & 127;
            float4 v = *(const float4*)(Lk + ((size_t)k * Nn + (n0 + nn)) * Nn + m0 + c4);
            float* dst = smem + BS0_OFF + (k * 128 + nn) * S + c4;
            dst[0] = v.x; dst[1] = v.y; dst[2] = v.z; dst[3] = v.w;
        }
        __syncthreads();
        compute_chunk(smem, XS0_OFF, BS0_OFF, tr2, tc2, l15, half, acc);
        __syncthreads();
    }
#endif

    // ---- spill Y to LDS (aliases Bs0): Ys[k][i][nn], stride SO
    {
        float* Ys = smem + BS0_OFF;
#pragma unroll
        for (int k = 0; k < KSu; ++k)
#pragma unroll
            for (int ri = 0; ri < 2; ++ri)
#pragma unroll
                for (int ci = 0; ci < 2; ++ci) {
                    int nn = (tc2 + ci) * 16 + l15;
#pragma unroll
                    for (int r = 0; r < 8; ++r) {
                        int i = (tr2 + ri) * 16 + r + 8 * half;
                        Ys[(k * 64 + i) * SO + nn] = acc[k][ri][ci][r];
                    }
                }
    }
    __syncthreads();

    // ---- GEMM2: Out[o, nn] = sum_k sum_i ThT[k][o][i] * Ys[k][i][nn]
    v8f oacc[2][2];
    {
        v8f z = {};
        oacc[0][0] = z; oacc[0][1] = z; oacc[1][0] = z; oacc[1][1] = z;
    }
#pragma unroll
    for (int k = 0; k < KSu; ++k) {
        const float* ThA = smem + TH_OFF + k * 64 * S;
        const float* Yk  = smem + BS0_OFF + k * 64 * SO;
        for (int kb = 0; kb < 64; kb += 4) {
            v2f a0 = *(const v2f*)(ThA + ((tr2 + 0) * 16 + l15) * S + kb + 2 * half);
            v2f a1 = *(const v2f*)(ThA + ((tr2 + 1) * 16 + l15) * S + kb + 2 * half);
            v2f b0, b1;
            int n0c = (tc2 + 0) * 16 + l15;
            int n1c = (tc2 + 1) * 16 + l15;
            b0[0] = Yk[(kb + 2 * half) * SO + n0c];
            b0[1] = Yk[(kb + 2 * half + 1) * SO + n0c];
            b1[0] = Yk[(kb + 2 * half) * SO + n1c];
            b1[1] = Yk[(kb + 2 * half + 1) * SO + n1c];
            oacc[0][0] = WMMA_F32(a0, b0, oacc[0][0]);
            oacc[0][1] = WMMA_F32(a0, b1, oacc[0][1]);
            oacc[1][0] = WMMA_F32(a1, b0, oacc[1][0]);
            oacc[1][1] = WMMA_F32(a1, b1, oacc[1][1]);
        }
    }

    // ---- epilogue: relu(out + bias + residual)
    if (!FUSE_FC) {
#pragma unroll
        for (int ri = 0; ri < 2; ++ri)
#pragma unroll
            for (int ci = 0; ci < 2; ++ci) {
                int nn = (tc2 + ci) * 16 + l15;
#pragma unroll
                for (int r = 0; r < 8; ++r) {
                    int o = (tr2 + ri) * 16 + r + 8 * half;
                    size_t idx = (((size_t)b * Cc + o) * Tt + t) * (size_t)Nn + n0 + nn;
                    float v = oacc[ri][ci][r] + bias[o] + x_in[idx];
                    out[idx] = fmaxf(v, 0.0f);
                }
            }
    } else {
        // x2 tile -> LDS (aliases Th region, dead after GEMM2)
        __syncthreads();
        float* Outs = smem + TH_OFF;
#pragma unroll
        for (int ri = 0; ri < 2; ++ri)
#pragma unroll
            for (int ci = 0; ci < 2; ++ci) {
                int nn = (tc2 + ci) * 16 + l15;
#pragma unroll
                for (int r = 0; r < 8; ++r) {
                    int o = (tr2 + ri) * 16 + r + 8 * half;
                    size_t idx = (((size_t)b * Cc + o) * Tt + t) * (size_t)Nn + n0 + nn;
                    float v = oacc[ri][ci][r] + bias[o] + x_in[idx];
                    Outs[o * SO + nn] = fmaxf(v, 0.0f);
                }
            }
        __syncthreads();

        // GEMM3: Final[nn, oo] = sum_c Outs[c][nn] * Fw[oo][c]
        const int nr2 = (wave >> 1) * 2;  // nn tiles
        const int oc2 = (wave & 1) * 2;   // oo tiles
        v8f facc[2][2];
        {
            v8f z = {};
            facc[0][0] = z; facc[0][1] = z; facc[1][0] = z; facc[1][1] = z;
        }
        const float* Fw = smem + FW_OFF;
        for (int kb = 0; kb < 64; kb += 4) {
            int m0c = (nr2 + 0) * 16 + l15;
            int m1c = (nr2 + 1) * 16 + l15;
            v2f a0, a1;
            a0[0] = Outs[(kb + 2 * half) * SO + m0c];
            a0[1] = Outs[(kb + 2 * half + 1) * SO + m0c];
            a1[0] = Outs[(kb + 2 * half) * SO + m1c];
            a1[1] = Outs[(kb + 2 * half + 1) * SO + m1c];
            v2f b0 = *(const v2f*)(Fw + ((oc2 + 0) * 16 + l15) * S + kb + 2 * half);
            v2f b1 = *(const v2f*)(Fw + ((oc2 + 1) * 16 + l15) * S + kb + 2 * half);
            facc[0][0] = WMMA_F32(a0, b0, facc[0][0]);
            facc[0][1] = WMMA_F32(a0, b1, facc[0][1]);
            facc[1][0] = WMMA_F32(a1, b0, facc[1][0]);
            facc[1][1] = WMMA_F32(a1, b1, facc[1][1]);
        }
#pragma unroll
        for (int ci = 0; ci < 2; ++ci) {
            int oo = (oc2 + ci) * 16 + l15;
            float bb = fc_b[oo];
#pragma unroll
            for (int ri = 0; ri < 2; ++ri)
#pragma unroll
                for (int r = 0; r < 8; ++r) {
                    int nn = (nr2 + ri) * 16 + r + 8 * half;
                    size_t idx = (((size_t)b * Tt + t) * (size_t)Nn + n0 + nn) * OUTc + oo;
                    out[idx] = facc[ri][ci][r] + bb;
                }
        }
    }
}

extern "C" void kernel_launch(void* const* d_in, const int* in_sizes, int n_in,
                              void* d_out, int out_size, void* d_ws, size_t ws_size,
                              hipStream_t stream) {
    const float* x      = (const float*)d_in[0];
    const float* Lk     = (const float*)d_in[1];
    const float* theta1 = (const float*)d_in[2];
    const float* b1     = (const float*)d_in[3];
    const float* theta2 = (const float*)d_in[4];
    const float* b2     = (const float*)d_in[5];
    const float* fc_w   = (const float*)d_in[6];
    const float* fc_b   = (const float*)d_in[7];
    float* out = (float*)d_out;
    float* x1  = (float*)d_ws;   // [B, C, T, N] intermediate = 100 MB

    dim3 grid(Nn / TN, Bsz * Tt);   // 8 x 384 workgroups
    size_t shmem = (size_t)LDS_FLOATS * sizeof(float);

    // Layer 1: x -> x1 (ws)
    gconv_layer_kernel<false><<<grid, 256, shmem, stream>>>(
        x, Lk, theta1, b1, nullptr, nullptr, x1);
    // Layer 2 + fused FC: x1 -> out
    gconv_layer_kernel<true><<<grid, 256, shmem, stream>>>(
        x1, Lk, theta2, b2, fc_w, fc_b, out);
}